// BatchCriterion_38843684225754
// MI455X (gfx1250) — compile-verified
//
#include <hip/hip_runtime.h>
#include <math.h>

#define D 128
#define KS 32                       // 128 / K=4
#define NS 8                        // column slices (parallelism: 64*NS workgroups)
#define EPS 1e-10f
#define EXP_C (1.4426950408889634f / 0.07f)   // log2(e)/T : exp(d/T) = 2^(d*EXP_C)

typedef float v2f __attribute__((ext_vector_type(2)));
typedef float v8f __attribute__((ext_vector_type(8)));

// reduce across the 16-lane half-wave (lanes 0-15 and 16-31 independently)
__device__ __forceinline__ float half_reduce(float v) {
    v += __shfl_xor(v, 1, 32);
    v += __shfl_xor(v, 2, 32);
    v += __shfl_xor(v, 4, 32);
    v += __shfl_xor(v, 8, 32);
    return v;
}

// One 16x16 tile of x * x^T with fp32 WMMA, K=128.
// Batched loads -> clauses + staged s_wait_loadcnt; two accumulator chains
// halve XDL RAW stalls.
__device__ __forceinline__ v8f tile_gemm(const v2f* __restrict__ a,
                                         const float* __restrict__ brow) {
    v2f b[KS];
#pragma unroll
    for (int ks = 0; ks < KS; ++ks)
        b[ks] = *(const v2f*)(brow + 4 * ks);
    v8f acc0 = {}, acc1 = {};
#pragma unroll
    for (int ks = 0; ks < KS; ks += 2) {
        acc0 = __builtin_amdgcn_wmma_f32_16x16x4_f32(
            false, a[ks],     false, b[ks],     (short)0, acc0, false, false);
        acc1 = __builtin_amdgcn_wmma_f32_16x16x4_f32(
            false, a[ks + 1], false, b[ks + 1], (short)0, acc1, false, false);
    }
    return acc0 + acc1;
}

// ---------------------------------------------------------------------------
// Pass 1: partial_div[slice*B + i] = sum_{j in slice, j != i} exp(sim[i][j]/T)
// grid = (B/128, NS); block = 256 (8 waves); wave owns rows [i0, i0+16).
// ---------------------------------------------------------------------------
__global__ __launch_bounds__(256) void kdiv(const float* __restrict__ x,
                                            float* __restrict__ partial, int B) {
    const int lane = threadIdx.x & 31;
    const int wave = threadIdx.x >> 5;
    const int m = lane & 15;         // M (or N) index within tile
    const int h = lane >> 4;         // K-half selector
    const int i0 = blockIdx.x * 128 + wave * 16;
    const int SL = B / NS;
    const int jbeg = blockIdx.y * SL, jend = jbeg + SL;

    // Preload A tile (16x128 fp32) into registers: k = 4*ks + 2h + v
    v2f a[KS];
    const float* arow = x + (size_t)(i0 + m) * D + 2 * h;
#pragma unroll
    for (int ks = 0; ks < KS; ++ks)
        a[ks] = *(const v2f*)(arow + 4 * ks);

    float rowsum[8];
#pragma unroll
    for (int v = 0; v < 8; ++v) rowsum[v] = 0.0f;

    for (int j0 = jbeg; j0 < jend; j0 += 16) {
        v8f acc = tile_gemm(a, x + (size_t)(j0 + m) * D + 2 * h);
        const int col = j0 + m;
#pragma unroll
        for (int v = 0; v < 8; ++v) {
            const int row = i0 + v + 8 * h;              // C layout: M = v + 8h
            float e = __builtin_amdgcn_exp2f(acc[v] * EXP_C);
            if (row == col) e = 0.0f;                    // zero diagonal
            rowsum[v] += e;
        }
    }
#pragma unroll
    for (int v = 0; v < 8; ++v) {
        float s = half_reduce(rowsum[v]);
        if (m == 0)                                      // unique owner per cell
            partial[(size_t)blockIdx.y * B + i0 + v + 8 * h] = s;
    }
}

// ---------------------------------------------------------------------------
// divs[i] = sum over slices of partial_div (deterministic fixed order)
// ---------------------------------------------------------------------------
__global__ __launch_bounds__(256) void kdivsum(const float* __restrict__ partial,
                                               float* __restrict__ divs, int B) {
    const int i = blockIdx.x * 256 + threadIdx.x;
    if (i >= B) return;
    float s = 0.0f;
#pragma unroll
    for (int sl = 0; sl < NS; ++sl) s += partial[(size_t)sl * B + i];
    divs[i] = s;
}

// ---------------------------------------------------------------------------
// Positive-pair term: row_term[i] = log(lnPmt) - log1p(-lnPmt); wave per row.
// ---------------------------------------------------------------------------
__global__ __launch_bounds__(256) void kpos(const float* __restrict__ x,
                                            const float* __restrict__ divs,
                                            float* __restrict__ row_term, int B) {
    const int lane = threadIdx.x & 31;
    const int wave = threadIdx.x >> 5;
    const int i = blockIdx.x * 8 + wave;
    if (i >= B) return;
    int p = i + (B >> 1); if (p >= B) p -= B;            // partner index
    const float* xi = x + (size_t)i * D;
    const float* xp = x + (size_t)p * D;
    float s = 0.0f;
#pragma unroll
    for (int q = 0; q < 4; ++q)
        s = fmaf(xi[lane + 32 * q], xp[lane + 32 * q], s);
    s += __shfl_xor(s, 1, 32);
    s += __shfl_xor(s, 2, 32);
    s += __shfl_xor(s, 4, 32);
    s += __shfl_xor(s, 8, 32);
    s += __shfl_xor(s, 16, 32);
    if (lane == 0) {
        float pos   = __builtin_amdgcn_exp2f(s * EXP_C);
        float lnPmt = pos / (divs[i] + EPS);
        row_term[i] = logf(lnPmt) - log1pf(-lnPmt);
    }
}

// ---------------------------------------------------------------------------
// Pass 2: partial_l1p[slice*B+i] = sum_{j in slice, j!=i}
//                                  log1p(-exp(sim[i][j]/T)/(div[i]+EPS))
// ---------------------------------------------------------------------------
__global__ __launch_bounds__(256) void kloss(const float* __restrict__ x,
                                             const float* __restrict__ divs,
                                             float* __restrict__ partial, int B) {
    const int lane = threadIdx.x & 31;
    const int wave = threadIdx.x >> 5;
    const int m = lane & 15;
    const int h = lane >> 4;
    const int i0 = blockIdx.x * 128 + wave * 16;
    const int SL = B / NS;
    const int jbeg = blockIdx.y * SL, jend = jbeg + SL;

    v2f a[KS];
    const float* arow = x + (size_t)(i0 + m) * D + 2 * h;
#pragma unroll
    for (int ks = 0; ks < KS; ++ks)
        a[ks] = *(const v2f*)(arow + 4 * ks);

    float inv[8], rowacc[8];
#pragma unroll
    for (int v = 0; v < 8; ++v) {
        inv[v] = 1.0f / (divs[i0 + v + 8 * h] + EPS);
        rowacc[v] = 0.0f;
    }

    for (int j0 = jbeg; j0 < jend; j0 += 16) {
        v8f acc = tile_gemm(a, x + (size_t)(j0 + m) * D + 2 * h);
        const int col = j0 + m;
#pragma unroll
        for (int v = 0; v < 8; ++v) {
            const int row = i0 + v + 8 * h;
            float e = __builtin_amdgcn_exp2f(acc[v] * EXP_C);
            if (row == col) e = 0.0f;                    // diag -> log1p(0) = 0
            rowacc[v] += log1pf(-(e * inv[v]));
        }
    }
#pragma unroll
    for (int v = 0; v < 8; ++v) {
        float s = half_reduce(rowacc[v]);
        if (m == 0)
            partial[(size_t)blockIdx.y * B + i0 + v + 8 * h] = s;
    }
}

// ---------------------------------------------------------------------------
// Final deterministic reduction:
// loss = -( sum_i row_term[i] + sum_i sum_s partial_l1p[s][i] ) / B
// ---------------------------------------------------------------------------
__global__ __launch_bounds__(256) void kfinal(const float* __restrict__ row_term,
                                              const float* __restrict__ partial,
                                              float* __restrict__ out, int B) {
    __shared__ float sm[256];
    float s = 0.0f;
    for (int i = threadIdx.x; i < B; i += 256) {         // fixed order per thread
        float r = row_term[i];
#pragma unroll
        for (int sl = 0; sl < NS; ++sl) r += partial[(size_t)sl * B + i];
        s += r;
    }
    sm[threadIdx.x] = s;
    __syncthreads();
    for (int off = 128; off > 0; off >>= 1) {
        if ((int)threadIdx.x < off) sm[threadIdx.x] += sm[threadIdx.x + off];
        __syncthreads();
    }
    if (threadIdx.x == 0) out[0] = -sm[0] / (float)B;
}

extern "C" void kernel_launch(void* const* d_in, const int* in_sizes, int n_in,
                              void* d_out, int out_size, void* d_ws, size_t ws_size,
                              hipStream_t stream) {
    (void)n_in; (void)out_size; (void)ws_size;
    const float* x = (const float*)d_in[0];
    const int B = in_sizes[0] / D;                 // 8192

    float* partial_div = (float*)d_ws;             // NS*B floats
    float* partial_l1p = partial_div + (size_t)NS * B;  // NS*B floats
    float* divs        = partial_l1p + (size_t)NS * B;  // B floats
    float* row_term    = divs + B;                 // B floats

    dim3 grid2(B / 128, NS);
    kdiv   <<<grid2,            256, 0, stream>>>(x, partial_div, B);
    kdivsum<<<(B + 255) / 256,  256, 0, stream>>>(partial_div, divs, B);
    kpos   <<<B / 8,            256, 0, stream>>>(x, divs, row_term, B);
    kloss  <<<grid2,            256, 0, stream>>>(x, divs, partial_l1p, B);
    kfinal <<<1,                256, 0, stream>>>(row_term, partial_l1p,
                                                 (float*)d_out, B);
}